// U2P_ODEFunc_18854906429535
// MI455X (gfx1250) — compile-verified
//
#include <hip/hip_runtime.h>

// Problem constants from the reference (fixed by setup_inputs()).
constexpr int D  = 128;     // emb dim
constexpr int Un = 100000;  // users (rows of y)

// ---------------------------------------------------------------------------
// Guaranteed-native no-return f32 atomic add (resolves in L2 atomic units).
// One 64-bit VGPR address pair, component selected by byte offset immediate.
// No-return atomics use STOREcnt; s_endpgm implicitly waits idle, so no
// explicit s_wait is required after these.
// ---------------------------------------------------------------------------
__device__ __forceinline__ void atomic_add4_f32_noret(float* p, float4 v) {
  asm volatile("global_atomic_add_f32 %0, %1, off"           :: "v"(p), "v"(v.x) : "memory");
  asm volatile("global_atomic_add_f32 %0, %1, off offset:4"  :: "v"(p), "v"(v.y) : "memory");
  asm volatile("global_atomic_add_f32 %0, %1, off offset:8"  :: "v"(p), "v"(v.z) : "memory");
  asm volatile("global_atomic_add_f32 %0, %1, off offset:12" :: "v"(p), "v"(v.w) : "memory");
}

// ---------------------------------------------------------------------------
// Zero the y scratch buffer (U x D floats). d_ws is poisoned once and not
// re-poisoned between replays, so this must run every call.
// ---------------------------------------------------------------------------
__global__ __launch_bounds__(256) void zero_f4(float4* __restrict__ p, int n4) {
  int i = blockIdx.x * blockDim.x + threadIdx.x;
  if (i < n4) p[i] = make_float4(0.f, 0.f, 0.f, 0.f);
}

// ---------------------------------------------------------------------------
// out = e - x   (epilogue pre-applied; SpMM2 then atomically adds z into out)
// ---------------------------------------------------------------------------
__global__ __launch_bounds__(256) void init_out_f4(const float4* __restrict__ x,
                                                   const float4* __restrict__ e,
                                                   float4* __restrict__ out, int n4) {
  int i = blockIdx.x * blockDim.x + threadIdx.x;
  if (i < n4) {
    float4 xv = x[i];
    float4 ev = e[i];
    out[i] = make_float4(ev.x - xv.x, ev.y - xv.y, ev.z - xv.z, ev.w - xv.w);
  }
}

// ---------------------------------------------------------------------------
// COO scatter SpMM: dst[rows[i], :] += vals[i] * src[cols[i], :]
// One wave32 per nonzero:
//   - nnz index forced wave-uniform (readfirstlane) so val/row/col fetches
//     become scalar s_load (KMcnt path), 3 scalars per 512B of useful work.
//   - 32 lanes x float4 = one fully coalesced 512B row gather
//     (global_load_b128), expected hit in the 192MB L2 (x=25.6MB, y=51.2MB).
//   - 4 native no-return global_atomic_add_f32 per lane via inline asm.
// ---------------------------------------------------------------------------
__global__ __launch_bounds__(256) void spmm_scatter(const float* __restrict__ vals,
                                                    const int* __restrict__ rows,
                                                    const int* __restrict__ cols,
                                                    const float* __restrict__ src,
                                                    float* __restrict__ dst,
                                                    int nnz) {
  // 8 waves per 256-thread block; wave-uniform nnz id.
  int i = __builtin_amdgcn_readfirstlane(blockIdx.x * 8 + (threadIdx.x >> 5));
  int lane = threadIdx.x & 31;
  if (i >= nnz) return;

  float v = vals[i];
  int   r = rows[i];
  int   c = cols[i];

  const float4* srow = (const float4*)(src + (long)c * D);
  float4 g = srow[lane];  // coalesced 512B row gather

  float4 contrib = make_float4(v * g.x, v * g.y, v * g.z, v * g.w);
  float* drow = dst + (long)r * D + lane * 4;
  atomic_add4_f32_noret(drow, contrib);
}

extern "C" void kernel_launch(void* const* d_in, const int* in_sizes, int n_in,
                              void* d_out, int out_size, void* d_ws, size_t ws_size,
                              hipStream_t stream) {
  // setup_inputs() order: t, x, e, vals_up, vals_pu, rows_up, cols_up, rows_pu, cols_pu
  const float* x       = (const float*)d_in[1];
  const float* e       = (const float*)d_in[2];
  const float* vals_up = (const float*)d_in[3];
  const float* vals_pu = (const float*)d_in[4];
  const int*   rows_up = (const int*)d_in[5];
  const int*   cols_up = (const int*)d_in[6];
  const int*   rows_pu = (const int*)d_in[7];
  const int*   cols_pu = (const int*)d_in[8];

  float* out = (float*)d_out;
  float* y   = (float*)d_ws;         // U x D floats = 51.2 MB scratch

  const int nnz = in_sizes[3];       // 1.6M
  const int yn4 = Un * D / 4;        // y float4 count
  const int on4 = out_size / 4;      // out float4 count (P*D/4)

  // 1) y = 0
  zero_f4<<<(yn4 + 255) / 256, 256, 0, stream>>>((float4*)y, yn4);

  // 2) out = e - x  (so SpMM2 can accumulate z straight into out)
  init_out_f4<<<(on4 + 255) / 256, 256, 0, stream>>>(
      (const float4*)x, (const float4*)e, (float4*)out, on4);

  // 3) y += HG_up @ x   (one wave per nnz, 8 waves/block)
  spmm_scatter<<<(nnz + 7) / 8, 256, 0, stream>>>(vals_up, rows_up, cols_up, x, y, nnz);

  // 4) out += HG_pu @ y
  spmm_scatter<<<(nnz + 7) / 8, 256, 0, stream>>>(vals_pu, rows_pu, cols_pu, y, out, nnz);
}